// ScaleLSD_65566970741124
// MI455X (gfx1250) — compile-verified
//
#include <hip/hip_runtime.h>
#include <math.h>

typedef __attribute__((ext_vector_type(2))) float v2f;
typedef __attribute__((ext_vector_type(8))) float v8f;

#define HH 256
#define WW 256
#define NPIX 65536
#define NB 2
#define NJ 512
#define JJ (NJ * NJ)
#define TH_J 0.008f
#define J2L_T 10.0f
#define SCALE_D 5.0f

// d_out flat layout (floats): lines_adj [2][J*J][4], counts [2][J*J], support_sum [2][J*J][4]
#define OUT_COUNTS  (NB * JJ * 4)              // 2097152
#define OUT_SUPPORT (OUT_COUNTS + NB * JJ)     // 2621440
#define OUT_ZERO_N  (NB * JJ * 5)              // counts + support contiguous region

// workspace float layout
#define WSF_LINES 0                            // NB*NPIX*4 floats
#define WSF_JLOC  (NB * NPIX * 4)              // NB*NPIX floats
#define WSF_A4    (WSF_JLOC + NB * NPIX)       // NB*NJ*4 floats: [-2x,-2y,x^2+y^2+pen,1]
#define WSF_JUNC  (WSF_A4 + NB * NJ * 4)       // NB*NJ*4 floats: [x,y,valid,score]
#define WSF_TOTAL (WSF_JUNC + NB * NJ * 4)
#define WSB_NCAND ((size_t)WSF_TOTAL * 4)      // byte offset: 2 ints
#define WSB_CAND  (WSB_NCAND + 16)             // byte offset: NB*NPIX ints

__device__ __forceinline__ float sigf(float x) { return 1.0f / (1.0f + expf(-x)); }

// ---------------- init: zero counts+support in d_out, default junction slots ----------------
__global__ void k_init(float* __restrict__ out, float* __restrict__ wf, int* __restrict__ ncand) {
    int i = blockIdx.x * 256 + threadIdx.x;
    if (i < OUT_ZERO_N) out[OUT_COUNTS + i] = 0.0f;
    if (i < NB * NJ) {
        float* A = wf + WSF_A4 + i * 4;
        A[0] = 0.0f; A[1] = 0.0f; A[2] = 1.0e12f; A[3] = 1.0f;   // invalid slot: pen=1e12
        float* JU = wf + WSF_JUNC + i * 4;
        JU[0] = 0.0f; JU[1] = 0.0f; JU[2] = 0.0f; JU[3] = 0.0f;
    }
    if (i < NB) ncand[i] = 0;
}

// ---------------- prep: decode lines + junction heatmap ----------------
__global__ void k_prep(const float* __restrict__ raw, float* __restrict__ wf) {
    int i = blockIdx.x * 256 + threadIdx.x;
    if (i >= NB * NPIX) return;
    int b = i >> 16;
    int p = i & (NPIX - 1);
    const float* R = raw + (size_t)b * 9 * NPIX;
    float md0 = sigf(R[0 * NPIX + p]);
    float md1 = sigf(R[1 * NPIX + p]);
    float md2 = sigf(R[2 * NPIX + p]);
    float dv  = sigf(R[3 * NPIX + p]);
    dv = fminf(fmaxf(dv, 0.0f), 1.0f);
    float md_un = (md0 - 0.5f) * 6.28318530717958647692f;
    float cs = cosf(md_un), ss = sinf(md_un);
    float yst = tanf(md1 * 1.57079632679489661923f);
    float yed = tanf(-md2 * 1.57079632679489661923f);
    float ds = dv * SCALE_D;
    float xst  = (cs - ss * yst) * ds;
    float yyst = (ss + cs * yst) * ds;
    float xed  = (cs - ss * yed) * ds;
    float yyed = (ss + cs * yed) * ds;
    float x0 = (float)(p & (WW - 1));
    float y0 = (float)(p >> 8);
    float4 ln;
    ln.x = fminf(fmaxf(xst  + x0, 0.0f), (float)(WW - 1));
    ln.y = fminf(fmaxf(yyst + y0, 0.0f), (float)(HH - 1));
    ln.z = fminf(fmaxf(xed  + x0, 0.0f), (float)(WW - 1));
    ln.w = fminf(fmaxf(yyed + y0, 0.0f), (float)(HH - 1));
    ((float4*)(wf + WSF_LINES))[i] = ln;
    // softmax over 2 channels, keep ch1 => sigmoid(r6 - r5)
    wf[WSF_JLOC + i] = sigf(R[6 * NPIX + p] - R[5 * NPIX + p]);
}

// ---------------- NMS + candidate compaction ----------------
__global__ void k_nms(const float* __restrict__ wf, int* __restrict__ ncand, int* __restrict__ cand) {
    int i = blockIdx.x * 256 + threadIdx.x;
    if (i >= NB * NPIX) return;
    int b = i >> 16;
    int p = i & (NPIX - 1);
    int x = p & (WW - 1), y = p >> 8;
    const float* JL = wf + WSF_JLOC + b * NPIX;
    float a = JL[p];
    float m = a;
    #pragma unroll
    for (int dy = -1; dy <= 1; ++dy)
        #pragma unroll
        for (int dx = -1; dx <= 1; ++dx) {
            int yy = y + dy, xx = x + dx;
            if (yy < 0 || yy >= HH || xx < 0 || xx >= WW) continue;
            m = fmaxf(m, JL[yy * WW + xx]);
        }
    if (a == m) {  // local maximum survives NMS (a > 0 always for sigmoid)
        int pos = atomicAdd(&ncand[b], 1);
        if (pos < NPIX) cand[b * NPIX + pos] = p;
    }
}

// ---------------- deterministic top-k by exact ranking (value desc, index asc) ----------------
__global__ __launch_bounds__(256) void k_rank(const float* __restrict__ raw, float* wf,
                                              const int* __restrict__ ncand,
                                              const int* __restrict__ cand) {
    __shared__ int   sIdx[256];
    __shared__ float sVal[256];
    const int b = blockIdx.y;
    const int tid = threadIdx.x;
    const int t = blockIdx.x * 256 + tid;
    const int nc = min(ncand[b], NPIX);
    const float* JL = wf + WSF_JLOC + b * NPIX;
    const int* CD = cand + b * NPIX;
    int myIdx = 0; float myV = 0.0f;
    const bool active = t < nc;
    if (active) { myIdx = CD[t]; myV = JL[myIdx]; }
    int rank = 0;
    for (int base = 0; base < nc; base += 256) {
        int i = base + tid;
        if (i < nc) { int ci = CD[i]; sIdx[tid] = ci; sVal[tid] = JL[ci]; }
        __syncthreads();
        int lim = min(256, nc - base);
        if (active) {
            for (int k = 0; k < lim; ++k) {
                float v = sVal[k]; int ci = sIdx[k];
                rank += ((v > myV) || (v == myV && ci < myIdx)) ? 1 : 0;
            }
        }
        __syncthreads();
    }
    if (active && rank < NJ) {
        const float* R = raw + (size_t)b * 9 * NPIX;
        float offx = sigf(R[7 * NPIX + myIdx]) - 0.5f;
        float offy = sigf(R[8 * NPIX + myIdx]) - 0.5f;
        float x = (float)(myIdx & (WW - 1)) + offx + 0.5f;
        float y = (float)(myIdx >> 8) + offy + 0.5f;
        bool valid = myV > TH_J;
        float* A = wf + WSF_A4 + (b * NJ + rank) * 4;
        A[0] = -2.0f * x; A[1] = -2.0f * y;
        A[2] = x * x + y * y + (valid ? 0.0f : 1.0e12f);
        A[3] = 1.0f;
        float* JU = wf + WSF_JUNC + (b * NJ + rank) * 4;
        JU[0] = x; JU[1] = y; JU[2] = valid ? 1.0f : 0.0f; JU[3] = myV;
    }
}

// ---------------- WMMA distance-field + argmin + bucket scatter ----------------
// dist(j, p) + pen_j = [-2jx, -2jy, |j|^2+pen, 1] . [px, py, 1, |p|^2]  (K = 4)
__global__ __launch_bounds__(256) void k_match(const float* __restrict__ wf, float* __restrict__ out) {
    __shared__ float sA[NJ * 4];
    __shared__ float sJ[NJ * 4];
    const int b = blockIdx.y;
    const int tid = threadIdx.x;
    for (int i = tid; i < NJ * 4; i += 256) {
        sA[i] = wf[WSF_A4 + b * NJ * 4 + i];
        sJ[i] = wf[WSF_JUNC + b * NJ * 4 + i];
    }
    __syncthreads();
    const int wave = tid >> 5;
    const int lane = tid & 31;
    const int n = lane & 15;
    const bool lo = lane < 16;
    const int lbase = blockIdx.x * 128 + wave * 16;
    const float4 ln = ((const float4*)(wf + WSF_LINES))[b * NPIX + lbase + n];
    // B layout (4x16 f32): VGPR0 = K0 | K2 across half-waves, VGPR1 = K1 | K3
    v2f bS, bE;
    bS.x = lo ? ln.x : 1.0f;
    bS.y = lo ? ln.y : (ln.x * ln.x + ln.y * ln.y);
    bE.x = lo ? ln.z : 1.0f;
    bE.y = lo ? ln.w : (ln.z * ln.z + ln.w * ln.w);
    float best1 = 3.0e38f, best2 = 3.0e38f;
    int idx1 = 0, idx2 = 0;
    const int koff  = lo ? 0 : 2;   // A layout: VGPR0 = K0|K2, VGPR1 = K1|K3
    const int rbase = lo ? 0 : 8;   // D layout: lanes 0-15 rows 0-7, lanes 16-31 rows 8-15
    for (int t = 0; t < 32; ++t) {
        const int abase = (t * 16 + n) * 4 + koff;
        v2f a;
        a.x = sA[abase];
        a.y = sA[abase + 1];
        v8f c1 = {};
        c1 = __builtin_amdgcn_wmma_f32_16x16x4_f32(false, a, false, bS, (short)0, c1, false, false);
        v8f c2 = {};
        c2 = __builtin_amdgcn_wmma_f32_16x16x4_f32(false, a, false, bE, (short)0, c2, false, false);
        #pragma unroll
        for (int v = 0; v < 8; ++v) {
            const int j = t * 16 + rbase + v;  // ascending scan -> strict '<' keeps first min
            float d1v = c1[v];
            if (d1v < best1) { best1 = d1v; idx1 = j; }
            float d2v = c2[v];
            if (d2v < best2) { best2 = d2v; idx2 = j; }
        }
    }
    // merge the two half-wave row classes with first-occurrence tie-break
    {
        float oV = __shfl_xor(best1, 16, 32); int oI = __shfl_xor(idx1, 16, 32);
        if (oV < best1 || (oV == best1 && oI < idx1)) { best1 = oV; idx1 = oI; }
        oV = __shfl_xor(best2, 16, 32); oI = __shfl_xor(idx2, 16, 32);
        if (oV < best2 || (oV == best2 && oI < idx2)) { best2 = oV; idx2 = oI; }
    }
    if (lo) {
        float j1x = sJ[idx1 * 4 + 0], j1y = sJ[idx1 * 4 + 1], v1 = sJ[idx1 * 4 + 2];
        float j2x = sJ[idx2 * 4 + 0], j2y = sJ[idx2 * 4 + 1], v2 = sJ[idx2 * 4 + 2];
        float dx = j1x - ln.x, dy = j1y - ln.y;
        float d1 = dx * dx + dy * dy;            // exact recompute for the threshold test
        dx = j2x - ln.z; dy = j2y - ln.w;
        float d2 = dx * dx + dy * dy;
        int imin = min(idx1, idx2), imax = max(idx1, idx2);
        bool keep = (imin < imax) && (d1 < J2L_T) && (d2 < J2L_T) && (v1 > 0.5f) && (v2 > 0.5f);
        if (keep) {
            int g = imin * NJ + imax;
            atomicAdd(out + OUT_COUNTS + b * JJ + g, 1.0f);
            float* sp = out + OUT_SUPPORT + (size_t)(b * JJ + g) * 4;
            atomicAdd(sp + 0, ln.x); atomicAdd(sp + 1, ln.y);
            atomicAdd(sp + 2, ln.z); atomicAdd(sp + 3, ln.w);
        }
    }
}

// ---------------- finalize: lines_adj from junction pairs where counts > 0 ----------------
__global__ void k_finalize(const float* __restrict__ wf, float* __restrict__ out) {
    int p = blockIdx.x * 256 + threadIdx.x;
    if (p >= NB * JJ) return;
    int b = p >> 18;
    int q = p & (JJ - 1);
    float cnt = out[OUT_COUNTS + p];
    float4 r = make_float4(0.0f, 0.0f, 0.0f, 0.0f);
    if (cnt > 0.0f) {
        int i = q >> 9, j = q & (NJ - 1);
        const float* JU = wf + WSF_JUNC + b * NJ * 4;
        r.x = JU[i * 4 + 0]; r.y = JU[i * 4 + 1];
        r.z = JU[j * 4 + 0]; r.w = JU[j * 4 + 1];
    }
    ((float4*)out)[p] = r;
}

extern "C" void kernel_launch(void* const* d_in, const int* in_sizes, int n_in,
                              void* d_out, int out_size, void* d_ws, size_t ws_size,
                              hipStream_t stream) {
    const float* raw = (const float*)d_in[0];
    float* out = (float*)d_out;
    float* wf = (float*)d_ws;
    int* ncand = (int*)((char*)d_ws + WSB_NCAND);
    int* cand  = (int*)((char*)d_ws + WSB_CAND);

    k_init    <<<dim3((OUT_ZERO_N + 255) / 256), 256, 0, stream>>>(out, wf, ncand);
    k_prep    <<<dim3((NB * NPIX + 255) / 256), 256, 0, stream>>>(raw, wf);
    k_nms     <<<dim3((NB * NPIX + 255) / 256), 256, 0, stream>>>(wf, ncand, cand);
    k_rank    <<<dim3(NPIX / 256, NB),          256, 0, stream>>>(raw, wf, ncand, cand);
    k_match   <<<dim3(NPIX / 128, NB),          256, 0, stream>>>(wf, out);
    k_finalize<<<dim3((NB * JJ + 255) / 256),   256, 0, stream>>>(wf, out);
}